// KANLayer_3513283248483
// MI455X (gfx1250) — compile-verified
//
#include <hip/hip_runtime.h>
#include <hip/hip_bf16.h>
#include <math.h>

// ---------------------------------------------------------------------------
// KAN layer  ==  fused GEMM:
//   out[b,o] = sum_{k'=i*9+f} feats[b,k'] * WeffT[o,k']  + biasC[o]
//   feats: on-the-fly basis expansion of x staged in LDS
//   WeffT[o][i*9+f] = W[i,o,f] * C[i,o]      (precomputed into d_ws)
//   biasC[o]        = sum_i bias[i,o]*C[i,o] (precomputed into d_ws)
// Matrix op: V_WMMA_F32_16X16X4_F32 (fp32-exact vs. the fp32 reference).
// ---------------------------------------------------------------------------

typedef float v2f __attribute__((ext_vector_type(2)));
typedef float v8f __attribute__((ext_vector_type(8)));

#define DIN        128
#define DOUT       128
#define NBASIS     9
#define KP         (DIN * NBASIS)   // 1152 = reduction length
#define ROWS       32               // rows of x per workgroup (2 M-tiles/wave)
#define LDS_STRIDE 1156             // 1152 + 4 pad dwords: stride%64==4 -> no bank conflicts
#define LDS_BYTES  (ROWS * LDS_STRIDE * 4)

// --------------------------- prep kernels ----------------------------------

__global__ __launch_bounds__(256) void kan_build_weff(
    const float* __restrict__ W,   // [DIN, DOUT, NBASIS]
    const float* __restrict__ C,   // [DIN, DOUT]
    float* __restrict__ WeffT)     // [DOUT, KP]
{
    int idx = blockIdx.x * 256 + threadIdx.x;       // over DOUT*KP = 147456
    if (idx >= DOUT * KP) return;
    int o = idx / KP;
    int r = idx - o * KP;
    int i = r / NBASIS;
    int f = r - i * NBASIS;
    WeffT[idx] = W[(i * DOUT + o) * NBASIS + f] * C[i * DOUT + o];
}

__global__ __launch_bounds__(128) void kan_build_biasc(
    const float* __restrict__ bias,  // [DIN, DOUT]
    const float* __restrict__ C,     // [DIN, DOUT]
    float* __restrict__ biasC)       // [DOUT]
{
    int o = threadIdx.x;
    float acc = 0.f;
    for (int i = 0; i < DIN; ++i)
        acc += bias[i * DOUT + o] * C[i * DOUT + o];
    biasC[o] = acc;
}

// --------------------------- main fused kernel -----------------------------

__global__ __launch_bounds__(256) void kan_main(
    const float* __restrict__ X,      // [B, DIN]
    const float* __restrict__ WeffT,  // [DOUT, KP]
    const float* __restrict__ biasC,  // [DOUT]
    float* __restrict__ out)          // [B, DOUT]
{
    extern __shared__ float lds[];            // ROWS x LDS_STRIDE feature tile

    const int tid = threadIdx.x;
    const int r0  = blockIdx.x * ROWS;

    // ---- Phase 1: basis expansion into LDS (32 rows x 1152 features) ----
    #pragma unroll
    for (int t = 0; t < (ROWS * DIN) / 256; ++t) {
        int e   = tid + t * 256;              // 0 .. 4095
        int row = e >> 7;                     // /DIN
        int i   = e & (DIN - 1);
        float xv = X[(size_t)(r0 + row) * DIN + i];
        float ax = fabsf(xv);
        float* p = &lds[row * LDS_STRIDE + i * NBASIS];
        p[0] = xv;
        p[1] = xv * xv;
        p[2] = xv * xv * xv;
        p[3] = expf(xv);
        p[4] = logf(ax + 1.0f);
        p[5] = sqrtf(ax);
        p[6] = tanhf(xv);
        p[7] = sinf(xv);
        p[8] = ax;
    }
    __syncthreads();

    // ---- Phase 2: WMMA GEMM over K'=1152, two 16x16 M-tiles per wave ----
    const int lane = tid & 31;
    const int wave = tid >> 5;
    const int n0   = wave * 16;               // N-tile of this wave
    const int rl   = lane & 15;               // row-in-tile (A) / col (B,D)
    const int koff = (lane >> 4) << 1;        // lanes 16-31 hold K+2,K+3

    // A operand: lane L<16 -> M=L,  {K=k0,k0+1}; lane L+16 -> M=L, {K=k0+2,k0+3}
    const float* a0p = &lds[rl * LDS_STRIDE + koff];
    const float* a1p = a0p + 16 * LDS_STRIDE;
    // B operand mirrors A with N in place of M; WeffT is [o][k'] so this is a b64 load
    const float* bp  = WeffT + (size_t)(n0 + rl) * KP + koff;

    v8f c0 = {};
    v8f c1 = {};

    #pragma unroll 4
    for (int k0 = 0; k0 < KP; k0 += 4) {
        v2f a0 = *(const v2f*)(a0p + k0);
        v2f a1 = *(const v2f*)(a1p + k0);
        v2f bb = *(const v2f*)(bp  + k0);
        // 8 args: (neg_a, A, neg_b, B, c_mod, C, reuse_a, reuse_b)
        c0 = __builtin_amdgcn_wmma_f32_16x16x4_f32(false, a0, false, bb,
                                                   (short)0, c0, false, false);
        c1 = __builtin_amdgcn_wmma_f32_16x16x4_f32(false, a1, false, bb,
                                                   (short)0, c1, false, false);
    }

    // ---- Phase 3: store D (VGPR j -> M = j + 8*(lane>=16), N = lane&15) ----
    const float bC   = biasC[n0 + rl];
    const int   mOff = (lane >> 4) * 8;
    #pragma unroll
    for (int j = 0; j < 8; ++j) {
        out[(size_t)(r0 + mOff + j)      * DOUT + n0 + rl] = c0[j] + bC;
        out[(size_t)(r0 + 16 + mOff + j) * DOUT + n0 + rl] = c1[j] + bC;
    }
}

// --------------------------- launcher --------------------------------------

extern "C" void kernel_launch(void* const* d_in, const int* in_sizes, int n_in,
                              void* d_out, int out_size, void* d_ws, size_t ws_size,
                              hipStream_t stream) {
    (void)n_in; (void)out_size; (void)ws_size;

    const float* x    = (const float*)d_in[0];  // [B, 128]
    const float* W    = (const float*)d_in[1];  // [128, 128, 9]
    const float* bias = (const float*)d_in[2];  // [128, 128]
    const float* C    = (const float*)d_in[3];  // [128, 128]
    float*       out  = (float*)d_out;          // [B, 128]

    float* WeffT = (float*)d_ws;                // KP*DOUT floats = 576 KiB
    float* biasC = WeffT + (size_t)KP * DOUT;   // + 128 floats

    const int B = in_sizes[0] / DIN;            // 4096

    kan_build_weff <<<(DOUT * KP + 255) / 256, 256, 0, stream>>>(W, C, WeffT);
    kan_build_biasc<<<1, 128, 0, stream>>>(bias, C, biasC);
    kan_main       <<<B / ROWS, 256, LDS_BYTES, stream>>>(x, WeffT, biasC, out);
}